// ClusteringHead_82025285419555
// MI455X (gfx1250) — compile-verified
//
#include <hip/hip_runtime.h>
#include <math.h>

typedef __attribute__((ext_vector_type(2))) float v2f;
typedef __attribute__((ext_vector_type(8))) float v8f;

#define D   128
#define H   128
#define HH2 64
#define KC  10
#define LN_EPS 1e-5f

// ---------------------------------------------------------------------------
// Kernel 1: three N x 128 = (N x 128) @ (128 x 128) fp32 GEMMs via
// V_WMMA_F32_16X16X4_F32.  One wave per (16-row stripe, matrix); the wave
// register-blocks all 8 column tiles (8 x v8f accumulators) so the A fragment
// is loaded once per K-step and feeds 8 WMMAs.
//   mat 0: h1pre = x @ w1
//   mat 1: hi    = x @ cw1[:D]
//   mat 2: hj    = x @ cw1[D:]
// ---------------------------------------------------------------------------
__global__ void wmma_gemm3_kernel(const float* __restrict__ x,
                                  const float* __restrict__ w1,
                                  const float* __restrict__ cw1,
                                  float* __restrict__ h1pre,
                                  float* __restrict__ hi,
                                  float* __restrict__ hj,
                                  int N) {
  const int rt  = blockIdx.x;          // 16-row stripe
  const int mat = blockIdx.y;          // 0,1,2

  const float* Bm  = (mat == 0) ? w1    : (mat == 1 ? cw1 : cw1 + D * H);
  float*       out = (mat == 0) ? h1pre : (mat == 1 ? hi  : hj);

  const int lane = threadIdx.x;        // 0..31 (wave32, EXEC all ones)
  const int half = lane >> 4;          // 0: K 0..1, 1: K 2..3
  const int l16  = lane & 15;

  int arow = rt * 16 + l16;
  if (arow >= N) arow = N - 1;         // clamp; masked on store
  const float* __restrict__ arow_p = x + (size_t)arow * D;

  v8f acc[8] = {};
#pragma unroll 4
  for (int k0 = 0; k0 < D; k0 += 4) {
    const int kk = k0 + half * 2;
    // A fragment (16x4): lane half selects K pair; one 8B load per K-step
    const v2f a = *(const v2f*)(arow_p + kk);
    // B fragments (4x16 per col tile), row-striped across lanes
    const float* __restrict__ brow0 = Bm + (size_t)kk * H + l16;
#pragma unroll
    for (int ct = 0; ct < 8; ++ct) {
      v2f b;
      b.x = brow0[ct * 16];
      b.y = brow0[ct * 16 + H];
      acc[ct] = __builtin_amdgcn_wmma_f32_16x16x4_f32(false, a, false, b,
                                                      (short)0, acc[ct],
                                                      false, false);
    }
  }

  // C layout: vgpr r -> M = r (lanes 0-15) / M = r + 8 (lanes 16-31)
  const int rbase = rt * 16 + half * 8;
  if (rt * 16 + 16 <= N) {             // full stripe: branch-free stores
#pragma unroll
    for (int ct = 0; ct < 8; ++ct)
#pragma unroll
      for (int r = 0; r < 8; ++r)
        out[(size_t)(rbase + r) * H + ct * 16 + l16] = acc[ct][r];
  } else {                             // tail stripe (rows >= N masked)
#pragma unroll
    for (int ct = 0; ct < 8; ++ct)
#pragma unroll
      for (int r = 0; r < 8; ++r)
        if (rbase + r < N)
          out[(size_t)(rbase + r) * H + ct * 16 + l16] = acc[ct][r];
  }
}

// ---------------------------------------------------------------------------
// Kernel 2: per-node cluster_net. One 128-thread block per node.
// h1 = relu(LN(h1pre + b1)); h2 = relu(LN(h1@w2 + b2)); logits = h2@w3 + b3;
// probs = softmax(logits)
// ---------------------------------------------------------------------------
__global__ void node_mlp_kernel(const float* __restrict__ h1pre,
                                const float* __restrict__ b1,
                                const float* __restrict__ g1,
                                const float* __restrict__ be1,
                                const float* __restrict__ w2,
                                const float* __restrict__ b2,
                                const float* __restrict__ g2,
                                const float* __restrict__ be2,
                                const float* __restrict__ w3,
                                const float* __restrict__ b3,
                                float* __restrict__ out_logits,
                                float* __restrict__ out_probs,
                                int N) {
  const int n = blockIdx.x;
  const int t = threadIdx.x;
  __shared__ float sA[128], sB[128];
  __shared__ float sh1[128];
  __shared__ float sh2[HH2];
  __shared__ float slog[KC];

  // ---- layer 1: LN(x@w1 + b1) + relu over H=128 ----
  float v = h1pre[(size_t)n * H + t] + b1[t];
  sA[t] = v;
  sB[t] = v * v;
  __syncthreads();
  for (int s = 64; s > 0; s >>= 1) {
    if (t < s) { sA[t] += sA[t + s]; sB[t] += sB[t + s]; }
    __syncthreads();
  }
  {
    float m   = sA[0] * (1.0f / 128.0f);
    float var = sB[0] * (1.0f / 128.0f) - m * m;
    float inv = rsqrtf(var + LN_EPS);
    float hv  = (v - m) * inv * g1[t] + be1[t];
    v = fmaxf(hv, 0.0f);
  }
  __syncthreads();               // all threads done reading sA[0]/sB[0]
  sh1[t] = v;
  __syncthreads();

  // ---- layer 2: h1@w2 + b2, LN over 64, relu ----
  float a2 = 0.0f;
  if (t < HH2) {
    a2 = b2[t];
    for (int k = 0; k < H; ++k) a2 += sh1[k] * w2[k * HH2 + t];
    sA[t] = a2;
    sB[t] = a2 * a2;
  }
  __syncthreads();
  for (int s = 32; s > 0; s >>= 1) {
    if (t < s) { sA[t] += sA[t + s]; sB[t] += sB[t + s]; }
    __syncthreads();
  }
  if (t < HH2) {
    float m2  = sA[0] * (1.0f / 64.0f);
    float v2  = sB[0] * (1.0f / 64.0f) - m2 * m2;
    float i2  = rsqrtf(v2 + LN_EPS);
    float h2v = (a2 - m2) * i2 * g2[t] + be2[t];
    sh2[t] = fmaxf(h2v, 0.0f);
  }
  __syncthreads();

  // ---- logits + softmax (TEMP = 1) ----
  if (t < KC) {
    float lg = b3[t];
    for (int k = 0; k < HH2; ++k) lg += sh2[k] * w3[k * KC + t];
    slog[t] = lg;
    out_logits[(size_t)n * KC + t] = lg;
  }
  __syncthreads();
  if (t < KC) {
    float mx = slog[0];
    for (int k = 1; k < KC; ++k) mx = fmaxf(mx, slog[k]);
    float ssum = 0.0f;
    for (int k = 0; k < KC; ++k) ssum += __expf(slog[k] - mx);
    out_probs[(size_t)n * KC + t] = __expf(slog[t] - mx) / ssum;
  }
}

// ---------------------------------------------------------------------------
// Kernel 3: cluster centers = (probs^T @ x) / (sum(probs,0) + 1e-8),
// plus passthrough of prototypes.  One block per cluster k, thread d.
// ---------------------------------------------------------------------------
__global__ void centers_kernel(const float* __restrict__ probs,
                               const float* __restrict__ x,
                               const float* __restrict__ prototypes,
                               float* __restrict__ out_centers,
                               float* __restrict__ out_protos,
                               int N) {
  const int k = blockIdx.x;
  const int d = threadIdx.x;
  __shared__ float sW[128];

  float wpart = 0.0f;
  for (int n = d; n < N; n += 128) wpart += probs[(size_t)n * KC + k];
  sW[d] = wpart;
  __syncthreads();
  for (int s = 64; s > 0; s >>= 1) {
    if (d < s) sW[d] += sW[d + s];
    __syncthreads();
  }
  const float wsum = sW[0] + 1e-8f;

  float acc = 0.0f;
  for (int n = 0; n < N; ++n)
    acc += probs[(size_t)n * KC + k] * x[(size_t)n * D + d];

  out_centers[k * D + d] = acc / wsum;
  out_protos[k * D + d]  = prototypes[k * D + d];
}

// ---------------------------------------------------------------------------
// Pair scoring primitive: wave32, 4 floats per lane (H=128).
// score(i,j) = tanh( relu(LN(hi[i]+hj[j]+cb1)) . cw2 + cb2 )
// ---------------------------------------------------------------------------
__device__ __forceinline__ float wave_sum32(float v) {
#pragma unroll
  for (int off = 16; off > 0; off >>= 1) v += __shfl_xor(v, off, 32);
  return v;
}

__device__ __forceinline__ float pair_score(const float4 hi4, const float4 hj4,
                                            const float4 cb, const float4 g,
                                            const float4 be, const float4 cw,
                                            float cb2) {
  float4 hp;
  hp.x = hi4.x + hj4.x + cb.x;
  hp.y = hi4.y + hj4.y + cb.y;
  hp.z = hi4.z + hj4.z + cb.z;
  hp.w = hi4.w + hj4.w + cb.w;

  float s  = hp.x + hp.y + hp.z + hp.w;
  float sq = hp.x * hp.x + hp.y * hp.y + hp.z * hp.z + hp.w * hp.w;
  s  = wave_sum32(s);
  sq = wave_sum32(sq);

  const float m   = s * (1.0f / 128.0f);
  const float var = sq * (1.0f / 128.0f) - m * m;
  const float inv = rsqrtf(var + LN_EPS);

  float dot = fmaxf((hp.x - m) * inv * g.x + be.x, 0.0f) * cw.x
            + fmaxf((hp.y - m) * inv * g.y + be.y, 0.0f) * cw.y
            + fmaxf((hp.z - m) * inv * g.z + be.z, 0.0f) * cw.z
            + fmaxf((hp.w - m) * inv * g.w + be.w, 0.0f) * cw.w;
  dot = wave_sum32(dot);
  return tanhf(dot + cb2);
}

// ---------------------------------------------------------------------------
// Kernel 4: dense comp_matrix[N,N].  One block per row i, 8 waves per block,
// one wave per pair; hi[i] + LN params hoisted into registers.  Next hj row
// stripe prefetched (global_prefetch_b8) behind the 3 wave reductions.
// ---------------------------------------------------------------------------
__global__ void comp_matrix_kernel(const float* __restrict__ hi,
                                   const float* __restrict__ hj,
                                   const float* __restrict__ cb1,
                                   const float* __restrict__ cg1,
                                   const float* __restrict__ cbe1,
                                   const float* __restrict__ cw2,
                                   const float* __restrict__ cb2p,
                                   float* __restrict__ out, int N) {
  const int i    = blockIdx.x;
  const int lane = threadIdx.x & 31;
  const int wid  = threadIdx.x >> 5;        // 0..7

  const float4 cb  = *(const float4*)(cb1  + 4 * lane);
  const float4 g   = *(const float4*)(cg1  + 4 * lane);
  const float4 be  = *(const float4*)(cbe1 + 4 * lane);
  const float4 cw  = *(const float4*)(cw2  + 4 * lane);
  const float  cb2 = cb2p[0];
  const float4 hi4 = *(const float4*)(hi + (size_t)i * H + 4 * lane);

  for (int j = wid; j < N; j += 8) {
    const float4 hj4 = *(const float4*)(hj + (size_t)j * H + 4 * lane);
    if (j + 8 < N)
      __builtin_prefetch(hj + (size_t)(j + 8) * H + 4 * lane, 0, 3);
    const float sc = pair_score(hi4, hj4, cb, g, be, cw, cb2);
    if (lane == 0) out[(size_t)i * N + j] = sc;
  }
}

// ---------------------------------------------------------------------------
// Kernel 5: comp_scores[E] on edges.  Grid-stride wave-per-edge; gathers hit L2
// (hi/hj total = 1 MB << 192 MB L2).
// ---------------------------------------------------------------------------
__global__ void comp_scores_kernel(const float* __restrict__ hi,
                                   const float* __restrict__ hj,
                                   const int* __restrict__ ei, int E,
                                   const float* __restrict__ cb1,
                                   const float* __restrict__ cg1,
                                   const float* __restrict__ cbe1,
                                   const float* __restrict__ cw2,
                                   const float* __restrict__ cb2p,
                                   float* __restrict__ out) {
  const int lane = threadIdx.x & 31;
  const int gw   = (int)((blockIdx.x * blockDim.x + threadIdx.x) >> 5);
  const int nw   = (int)((gridDim.x * blockDim.x) >> 5);

  const float4 cb  = *(const float4*)(cb1  + 4 * lane);
  const float4 g   = *(const float4*)(cg1  + 4 * lane);
  const float4 be  = *(const float4*)(cbe1 + 4 * lane);
  const float4 cw  = *(const float4*)(cw2  + 4 * lane);
  const float  cb2 = cb2p[0];

  for (int e = gw; e < E; e += nw) {
    const int i = ei[e];
    const int j = ei[E + e];
    const float4 hi4 = *(const float4*)(hi + (size_t)i * H + 4 * lane);
    const float4 hj4 = *(const float4*)(hj + (size_t)j * H + 4 * lane);
    const float sc = pair_score(hi4, hj4, cb, g, be, cw, cb2);
    if (lane == 0) out[e] = sc;
  }
}

// ---------------------------------------------------------------------------
extern "C" void kernel_launch(void* const* d_in, const int* in_sizes, int n_in,
                              void* d_out, int out_size, void* d_ws, size_t ws_size,
                              hipStream_t stream) {
  const float* x    = (const float*)d_in[0];
  const int*   ei   = (const int*)  d_in[1];
  const float* w1   = (const float*)d_in[2];
  const float* b1   = (const float*)d_in[3];
  const float* g1   = (const float*)d_in[4];
  const float* be1  = (const float*)d_in[5];
  const float* w2   = (const float*)d_in[6];
  const float* b2   = (const float*)d_in[7];
  const float* g2   = (const float*)d_in[8];
  const float* be2  = (const float*)d_in[9];
  const float* w3   = (const float*)d_in[10];
  const float* b3   = (const float*)d_in[11];
  const float* cw1  = (const float*)d_in[12];
  const float* cb1  = (const float*)d_in[13];
  const float* cg1  = (const float*)d_in[14];
  const float* cbe1 = (const float*)d_in[15];
  const float* cw2  = (const float*)d_in[16];
  const float* cb2  = (const float*)d_in[17];
  const float* prot = (const float*)d_in[18];

  const int N = in_sizes[0] / D;     // 1000
  const int E = in_sizes[1] / 2;     // 500000

  // workspace: h1pre | hi | hj  (3 * N * 128 floats = 1.5 MB)
  float* h1pre = (float*)d_ws;
  float* hi    = h1pre + (size_t)N * H;
  float* hj    = hi    + (size_t)N * H;

  // output layout (return-order concat)
  float* out       = (float*)d_out;
  float* o_logits  = out;
  float* o_probs   = o_logits  + (size_t)N * KC;
  float* o_scores  = o_probs   + (size_t)N * KC;
  float* o_matrix  = o_scores  + (size_t)E;
  float* o_centers = o_matrix  + (size_t)N * N;
  float* o_protos  = o_centers + (size_t)KC * D;

  const int rowTiles = (N + 15) / 16;

  wmma_gemm3_kernel<<<dim3(rowTiles, 3), 32, 0, stream>>>(
      x, w1, cw1, h1pre, hi, hj, N);

  node_mlp_kernel<<<N, 128, 0, stream>>>(
      h1pre, b1, g1, be1, w2, b2, g2, be2, w3, b3, o_logits, o_probs, N);

  centers_kernel<<<KC, 128, 0, stream>>>(
      o_probs, x, prot, o_centers, o_protos, N);

  comp_matrix_kernel<<<N, 256, 0, stream>>>(
      hi, hj, cb1, cg1, cbe1, cw2, cb2, o_matrix, N);

  comp_scores_kernel<<<2048, 256, 0, stream>>>(
      hi, hj, ei, E, cb1, cg1, cbe1, cw2, cb2, o_scores);
}